// domain_model_75033078661527
// MI455X (gfx1250) — compile-verified
//
#include <hip/hip_runtime.h>

#define B_SZ   2048
#define D_SZ   512
#define K_HIST 200
#define CB_SZ  2048
#define HID_SZ 1024
#define NTILE_CB (CB_SZ / 16)   // 128 codebook tiles per row

typedef __attribute__((ext_vector_type(2))) float v2f;
typedef __attribute__((ext_vector_type(8))) float v8f;

// ---------------------------------------------------------------------------
// 1) Masked mean over interaction history: mean_out[b] = sum(E[items])/count
// ---------------------------------------------------------------------------
__global__ void mean_hist_kernel(const int* __restrict__ items,
                                 const float* __restrict__ item_embed,
                                 float* __restrict__ mean_out) {
    int b = blockIdx.x;
    int t = threadIdx.x;                 // 128 threads, 4 floats each (D=512)
    const int* row_items = items + (size_t)b * K_HIST;
    float ax = 0.f, ay = 0.f, az = 0.f, aw = 0.f, cnt = 0.f;
    for (int k = 0; k < K_HIST; ++k) {
        int it = row_items[k];
        if (it != 0) {                   // PAD == 0
            const float4* src = (const float4*)(item_embed + (size_t)it * D_SZ);
            float4 v = src[t];
            ax += v.x; ay += v.y; az += v.z; aw += v.w;
            cnt += 1.f;
        }
    }
    float inv = 1.0f / cnt;
    float4 o; o.x = ax * inv; o.y = ay * inv; o.z = az * inv; o.w = aw * inv;
    ((float4*)(mean_out + (size_t)b * D_SZ))[t] = o;
}

// ---------------------------------------------------------------------------
// 2) pos/neg row gathers straight into d_out
// ---------------------------------------------------------------------------
__global__ void gather_pn_kernel(const int* __restrict__ pos,
                                 const int* __restrict__ neg,
                                 const float* __restrict__ item_embed,
                                 float* __restrict__ out_pos,
                                 float* __restrict__ out_neg) {
    int b = blockIdx.x;
    int t = threadIdx.x;                 // 128 threads
    const float4* ps = (const float4*)(item_embed + (size_t)pos[b] * D_SZ);
    const float4* ns = (const float4*)(item_embed + (size_t)neg[b] * D_SZ);
    ((float4*)(out_pos + (size_t)b * D_SZ))[t] = ps[t];
    ((float4*)(out_neg + (size_t)b * D_SZ))[t] = ns[t];
}

// ---------------------------------------------------------------------------
// 3) Codebook squared norms
// ---------------------------------------------------------------------------
__global__ void cnorm_kernel(const float* __restrict__ cb,
                             float* __restrict__ cnorm) {
    __shared__ float sh[128];
    int c = blockIdx.x;
    int t = threadIdx.x;                 // 128 threads * float4 = 512
    const float4* row = (const float4*)(cb + (size_t)c * D_SZ);
    float4 v = row[t];
    sh[t] = v.x * v.x + v.y * v.y + v.z * v.z + v.w * v.w;
    __syncthreads();
    for (int s = 64; s > 0; s >>= 1) {
        if (t < s) sh[t] += sh[t + s];
        __syncthreads();
    }
    if (t == 0) cnorm[c] = sh[0];
}

// ---------------------------------------------------------------------------
// 4) f32 WMMA GEMM: C = act(A[M,K] @ Bm[K,N] + bias), one 16x16 tile per wave
//    Uses V_WMMA_F32_16X16X4_F32, K stepped by 4 per the ISA f32 layouts.
// ---------------------------------------------------------------------------
__global__ void wmma_gemm_kernel(const float* __restrict__ A,
                                 const float* __restrict__ Bm,
                                 const float* __restrict__ bias,
                                 float* __restrict__ C,
                                 float* __restrict__ Cdup,
                                 int M, int K, int N, int relu) {
    int wid  = (blockIdx.x * blockDim.x + threadIdx.x) >> 5;
    int lane = threadIdx.x & 31;
    int ntiles = N >> 4;
    int mt = wid / ntiles;
    int nt = wid - mt * ntiles;
    int m0 = mt << 4, n0 = nt << 4;
    int half = lane >> 4, l16 = lane & 15;

    const float* arow = A + (size_t)(m0 + l16) * K;   // A: M = l16 (+tile)
    v8f acc = {};
    for (int k = 0; k < K; k += 4) {
        int ka = k + 2 * half;                        // K split across lane halves
        v2f a; a.x = arow[ka];
               a.y = arow[ka + 1];
        v2f b; b.x = Bm[(size_t)ka * N + n0 + l16];   // B: N = l16, rows = K
               b.y = Bm[(size_t)(ka + 1) * N + n0 + l16];
        acc = __builtin_amdgcn_wmma_f32_16x16x4_f32(
            false, a, false, b, (short)0, acc, false, false);
    }
    float bv = bias[n0 + l16];
#pragma unroll
    for (int r = 0; r < 8; ++r) {                     // D: M = r + 8*half, N = l16
        float v = acc[r] + bv;
        if (relu) v = fmaxf(v, 0.f);
        size_t idx = (size_t)(m0 + r + 8 * half) * N + (n0 + l16);
        C[idx] = v;
        if (Cdup) Cdup[idx] = v;
    }
}

// ---------------------------------------------------------------------------
// 5) VQ scores via WMMA (S = U @ CBᵀ), fused per-tile argmin of (‖c‖² - 2S)
//    Butterfly shuffle over the 16 N-lanes; lane 0/16 emit per-row partials.
// ---------------------------------------------------------------------------
__global__ void vq_tile_kernel(const float* __restrict__ U,
                               const float* __restrict__ cb,
                               const float* __restrict__ cnorm,
                               float* __restrict__ pval,
                               int* __restrict__ pidx) {
    int wid  = (blockIdx.x * blockDim.x + threadIdx.x) >> 5;
    int lane = threadIdx.x & 31;
    int mt = wid / NTILE_CB;
    int nt = wid - mt * NTILE_CB;
    int m0 = mt << 4, n0 = nt << 4;
    int half = lane >> 4, l16 = lane & 15;

    const float* arow = U  + (size_t)(m0 + l16) * D_SZ;
    const float* brow = cb + (size_t)(n0 + l16) * D_SZ; // Bᵀ[k][n] = cb[n][k]
    v8f acc = {};
    for (int k = 0; k < D_SZ; k += 4) {
        int ka = k + 2 * half;
        v2f a; a.x = arow[ka]; a.y = arow[ka + 1];
        v2f b; b.x = brow[ka]; b.y = brow[ka + 1];
        acc = __builtin_amdgcn_wmma_f32_16x16x4_f32(
            false, a, false, b, (short)0, acc, false, false);
    }
    float cn = cnorm[n0 + l16];
#pragma unroll
    for (int r = 0; r < 8; ++r) {
        float v = cn - 2.0f * acc[r];                 // ‖u‖² dropped (row-const)
        int idx = n0 + l16;
#pragma unroll
        for (int off = 1; off < 16; off <<= 1) {      // reduce across N lanes
            float ov = __shfl_xor(v, off, 32);
            int   oi = __shfl_xor(idx, off, 32);
            if (ov < v || (ov == v && oi < idx)) { v = ov; idx = oi; }
        }
        if (l16 == 0) {
            int row = m0 + r + 8 * half;
            pval[(size_t)row * NTILE_CB + nt] = v;
            pidx[(size_t)row * NTILE_CB + nt] = idx;
        }
    }
}

// ---------------------------------------------------------------------------
// 6) Per-row final argmin + quant_user = codebook[id] + diff accumulation
// ---------------------------------------------------------------------------
__global__ void argmin_row_kernel(const float* __restrict__ pval,
                                  const int* __restrict__ pidx,
                                  const float* __restrict__ U,
                                  const float* __restrict__ cb,
                                  float* __restrict__ quant_out,
                                  float* __restrict__ diff_acc) {
    __shared__ float sv[128];
    __shared__ int   si[128];
    int b = blockIdx.x;
    int t = threadIdx.x;                 // 128 == NTILE_CB
    sv[t] = pval[(size_t)b * NTILE_CB + t];
    si[t] = pidx[(size_t)b * NTILE_CB + t];
    __syncthreads();
    for (int s = 64; s > 0; s >>= 1) {
        if (t < s) {
            float ov = sv[t + s]; int oi = si[t + s];
            if (ov < sv[t] || (ov == sv[t] && oi < si[t])) { sv[t] = ov; si[t] = oi; }
        }
        __syncthreads();
    }
    int best = si[0];
    __syncthreads();

    const float* q = cb + (size_t)best * D_SZ;
    const float* u = U  + (size_t)b * D_SZ;
    float ssum = 0.f;
    for (int j = t; j < D_SZ; j += 128) {
        float qq = q[j];
        quant_out[(size_t)b * D_SZ + j] = qq;         // straight-through == q
        float dd = qq - u[j];
        ssum += dd * dd;
    }
    sv[t] = ssum;
    __syncthreads();
    for (int s = 64; s > 0; s >>= 1) {
        if (t < s) sv[t] += sv[t + s];
        __syncthreads();
    }
    if (t == 0) atomicAdd(diff_acc, sv[0]);
}

__global__ void zero_scalar_kernel(float* p) { *p = 0.f; }

__global__ void finalize_kernel(const float* __restrict__ acc,
                                float* __restrict__ out) {
    *out = *acc * (1.0f / (float)((size_t)B_SZ * D_SZ));
}

// ---------------------------------------------------------------------------
extern "C" void kernel_launch(void* const* d_in, const int* in_sizes, int n_in,
                              void* d_out, int out_size, void* d_ws, size_t ws_size,
                              hipStream_t stream) {
    (void)in_sizes; (void)n_in; (void)out_size; (void)ws_size;
    const int*   items      = (const int*)d_in[1];
    const int*   pos        = (const int*)d_in[2];
    const int*   neg        = (const int*)d_in[3];
    const float* item_embed = (const float*)d_in[4];
    const float* W1         = (const float*)d_in[5];
    const float* b1         = (const float*)d_in[6];
    const float* W2         = (const float*)d_in[7];
    const float* b2         = (const float*)d_in[8];
    const float* codebook   = (const float*)d_in[9];

    float* out = (float*)d_out;
    float* out_quant = out;                                   // [B,D]
    float* out_pos   = out + (size_t)B_SZ * D_SZ;             // [B,D]
    float* out_neg   = out + (size_t)2 * B_SZ * D_SZ;         // [B,D]
    float* out_diff  = out + (size_t)3 * B_SZ * D_SZ;         // scalar
    float* out_uemb  = out_diff + 1;                          // [B,D]

    // workspace layout (bytes); mean buffer is reused for user_embed after gemm1
    char* ws = (char*)d_ws;
    float* ws_meanU = (float*)(ws);                                   // 4 MB
    float* ws_H     = (float*)(ws + 4194304);                         // 8 MB
    float* ws_cnorm = (float*)(ws + 12582912);                        // 8 KB
    float* ws_pval  = (float*)(ws + 12591104);                        // 1 MB
    int*   ws_pidx  = (int*)  (ws + 13639680);                        // 1 MB
    float* ws_diff  = (float*)(ws + 14688256);                        // 4 B

    zero_scalar_kernel<<<1, 1, 0, stream>>>(ws_diff);
    mean_hist_kernel<<<B_SZ, 128, 0, stream>>>(items, item_embed, ws_meanU);
    gather_pn_kernel<<<B_SZ, 128, 0, stream>>>(pos, neg, item_embed, out_pos, out_neg);
    cnorm_kernel<<<CB_SZ, 128, 0, stream>>>(codebook, ws_cnorm);

    // H = relu(mean @ W1 + b1): (2048/16)*(1024/16) = 8192 waves -> 2048 blocks
    wmma_gemm_kernel<<<2048, 128, 0, stream>>>(
        ws_meanU, W1, b1, ws_H, (float*)nullptr, B_SZ, D_SZ, HID_SZ, 1);

    // U = H @ W2 + b2: (2048/16)*(512/16) = 4096 waves -> 1024 blocks
    // writes aligned ws copy (overwriting mean, now dead) + d_out user_embed
    wmma_gemm_kernel<<<1024, 128, 0, stream>>>(
        ws_H, W2, b2, ws_meanU, out_uemb, B_SZ, HID_SZ, D_SZ, 0);

    // VQ scores + per-tile argmin: 128*128 = 16384 waves -> 4096 blocks
    vq_tile_kernel<<<4096, 128, 0, stream>>>(
        ws_meanU, codebook, ws_cnorm, ws_pval, ws_pidx);

    argmin_row_kernel<<<B_SZ, 128, 0, stream>>>(
        ws_pval, ws_pidx, ws_meanU, codebook, out_quant, ws_diff);

    finalize_kernel<<<1, 1, 0, stream>>>(ws_diff, out_diff);
}